// CascadeHierarchicalEmbedding_46703474377118
// MI455X (gfx1250) — compile-verified
//
#include <hip/hip_runtime.h>
#include <math.h>

// CascadeHierarchicalEmbedding for MI455X (gfx1250, wave32).
// Gather-bandwidth-bound problem (~850MB @ 23.3TB/s); gate MLPs done with
// V_WMMA_F32_16X16X4_F32 (fp32) since compute is free vs. HBM gathers.

typedef __attribute__((ext_vector_type(2))) float v2f;
typedef __attribute__((ext_vector_type(4))) float v4f;
typedef __attribute__((ext_vector_type(8))) float v8f;

#define DIM     64
#define GATE_H  32
#define WAVES   4          // waves (tiles) per block
#define XSTR    132        // padded LDS row stride in floats (16B-aligned, bank-conflict-free)

__device__ __forceinline__ float fast_sigmoid(float x) {
    return 1.0f / (1.0f + __expf(-x));   // v_exp_f32
}

// Evaluate one gate MLP for the wave's 16-token tile staged in xbuf (16 x 128, stride XSTR).
// Returns this lane's token gate scalar (token = lane>>1).
__device__ __forceinline__ float gate_eval(const float* __restrict__ xbuf,
                                           const float* __restrict__ w1,  // [128][32] row-major
                                           const float* __restrict__ b1,  // [32]
                                           const float* __restrict__ w2,  // [32]
                                           float b2v, int lane)
{
    v8f c0 = {0.f,0.f,0.f,0.f,0.f,0.f,0.f,0.f};
    v8f c1 = {0.f,0.f,0.f,0.f,0.f,0.f,0.f,0.f};

    const int Lm  = lane & 15;
    const int khi = (lane & 16) >> 3;              // 0 for lanes 0-15, 2 for lanes 16-31
    const float* arow = xbuf + Lm * XSTR + khi;    // A: lane Lm holds row M=Lm, K = k+khi .. k+khi+1

    #pragma unroll
    for (int k = 0; k < 2 * DIM; k += 4) {
        v2f a = *(const v2f*)(arow + k);           // ds_load_b64, conflict-free (stride 132)
        const int kk = k + khi;
        // B fragments (4x16 KxN): lane holds B[kk][n], B[kk+1][n]
        v2f bA, bB;
        bA.x = w1[(kk    ) * GATE_H + Lm];
        bA.y = w1[(kk + 1) * GATE_H + Lm];
        bB.x = w1[(kk    ) * GATE_H + 16 + Lm];
        bB.y = w1[(kk + 1) * GATE_H + 16 + Lm];
        c0 = __builtin_amdgcn_wmma_f32_16x16x4_f32(false, a, false, bA, (short)0, c0, false, false);
        c1 = __builtin_amdgcn_wmma_f32_16x16x4_f32(false, a, false, bB, (short)0, c1, false, false);
    }

    // bias + ReLU, then fold the 32->1 layer into a per-lane partial dot with w2.
    const float bb0 = b1[Lm],      bb1 = b1[16 + Lm];
    const float wa  = w2[Lm],      wb  = w2[16 + Lm];
    float part[8];
    #pragma unroll
    for (int v = 0; v < 8; ++v) {
        float h0 = fmaxf(c0[v] + bb0, 0.0f);       // N = Lm
        float h1 = fmaxf(c1[v] + bb1, 0.0f);       // N = 16 + Lm
        part[v] = h0 * wa + h1 * wb;               // partial over this lane's two N values
    }
    // Reduce over N: 16-lane xor-butterfly within each half (M = v for lanes 0-15, v+8 for 16-31).
    #pragma unroll
    for (int v = 0; v < 8; ++v) {
        float s = part[v];
        #pragma unroll
        for (int off = 1; off < 16; off <<= 1)
            s += __shfl_xor(s, off, 32);
        part[v] = s;
    }
    // Lane handles token t = lane>>1; C-layout puts s[t] in part[t & 7] on this lane's half.
    const int sel = (lane >> 1) & 7;
    float s = part[0];
    #pragma unroll
    for (int v = 1; v < 8; ++v)
        if (sel == v) s = part[v];
    return fast_sigmoid(s + b2v);
}

__global__ __launch_bounds__(WAVES * 32)
void cascade_embed_kernel(const int* __restrict__ ids0, const int* __restrict__ ids1,
                          const int* __restrict__ ids2,
                          const float* __restrict__ emb0, const float* __restrict__ emb1,
                          const float* __restrict__ emb2,
                          const float* __restrict__ g0w1, const float* __restrict__ g0b1,
                          const float* __restrict__ g0w2, const float* __restrict__ g0b2,
                          const float* __restrict__ g1w1, const float* __restrict__ g1b1,
                          const float* __restrict__ g1w2, const float* __restrict__ g1b2,
                          float* __restrict__ out, int ntiles)
{
    __shared__ float s_x[WAVES][16 * XSTR];        // 33,792 B

    const int lane = threadIdx.x & 31;
    const int wave = threadIdx.x >> 5;
    float* xbuf = s_x[wave];

    int tile = blockIdx.x * WAVES + wave;
    if (tile >= ntiles) tile = ntiles - 1;         // keep waves full: WMMA needs EXEC all-1s,
                                                   // and barriers must be reached uniformly
    const int  t = lane >> 1;                      // token slot 0..15
    const int  j = lane & 1;                       // which 32-float half of the 64-dim row
    const long n = (long)tile * 16 + t;            // global token

    const int i0 = ids0[n], i1 = ids1[n], i2 = ids2[n];

    v4f fine[8], cur[8];
    {   // coarsest level: cur = emb2[ids2]
        const v4f* src = (const v4f*)(emb2 + (size_t)i2 * DIM + j * 32);
        #pragma unroll
        for (int i = 0; i < 8; ++i) cur[i] = src[i];      // global_load_b128 x8
    }
    {   // level-1 fine = emb1[ids1]
        const v4f* src = (const v4f*)(emb1 + (size_t)i1 * DIM + j * 32);
        #pragma unroll
        for (int i = 0; i < 8; ++i) fine[i] = src[i];
    }
    {   // stage x = [fine1 | cur2] into this wave's LDS tile
        v4f* dF = (v4f*)(xbuf + t * XSTR + j * 32);
        v4f* dC = (v4f*)(xbuf + t * XSTR + DIM + j * 32);
        #pragma unroll
        for (int i = 0; i < 8; ++i) { dF[i] = fine[i]; dC[i] = cur[i]; }
    }
    __syncthreads();

    // gate 1 + blend (registers)
    const float g1 = gate_eval(xbuf, g1w1, g1b1, g1w2, g1b2[0], lane);
    #pragma unroll
    for (int i = 0; i < 8; ++i) cur[i] = g1 * fine[i] + (1.0f - g1) * cur[i];

    // level-0 fine = emb0[ids0]  (256MB table -> true HBM gathers)
    {
        const v4f* src = (const v4f*)(emb0 + (size_t)i0 * DIM + j * 32);
        #pragma unroll
        for (int i = 0; i < 8; ++i) fine[i] = src[i];
    }
    __syncthreads();                                // retire gate-1 LDS reads before overwrite
    {   // restage x = [fine0 | cur]
        v4f* dF = (v4f*)(xbuf + t * XSTR + j * 32);
        v4f* dC = (v4f*)(xbuf + t * XSTR + DIM + j * 32);
        #pragma unroll
        for (int i = 0; i < 8; ++i) { dF[i] = fine[i]; dC[i] = cur[i]; }
    }
    __syncthreads();

    // gate 0 + blend
    const float g0 = gate_eval(xbuf, g0w1, g0b1, g0w2, g0b2[0], lane);
    #pragma unroll
    for (int i = 0; i < 8; ++i) cur[i] = g0 * fine[i] + (1.0f - g0) * cur[i];

    // streaming output: 256B/token, never re-read -> nontemporal b128 stores
    float* dst = out + (size_t)n * DIM + j * 32;
    #pragma unroll
    for (int i = 0; i < 8; ++i)
        __builtin_nontemporal_store(cur[i], (v4f*)(dst + i * 4));
}

extern "C" void kernel_launch(void* const* d_in, const int* in_sizes, int n_in,
                              void* d_out, int out_size, void* d_ws, size_t ws_size,
                              hipStream_t stream)
{
    const int*   ids0 = (const int*)  d_in[0];
    const int*   ids1 = (const int*)  d_in[1];
    const int*   ids2 = (const int*)  d_in[2];
    const float* emb0 = (const float*)d_in[3];
    const float* emb1 = (const float*)d_in[4];
    const float* emb2 = (const float*)d_in[5];
    const float* g0w1 = (const float*)d_in[6];
    const float* g0b1 = (const float*)d_in[7];
    const float* g0w2 = (const float*)d_in[8];
    const float* g0b2 = (const float*)d_in[9];
    const float* g1w1 = (const float*)d_in[10];
    const float* g1b1 = (const float*)d_in[11];
    const float* g1w2 = (const float*)d_in[12];
    const float* g1b2 = (const float*)d_in[13];
    float* out = (float*)d_out;

    const int ntok   = in_sizes[0];                 // B*H = 819200
    const int ntiles = (ntok + 15) / 16;            // 51200
    const int nblk   = (ntiles + WAVES - 1) / WAVES;

    cascade_embed_kernel<<<nblk, WAVES * 32, 0, stream>>>(
        ids0, ids1, ids2, emb0, emb1, emb2,
        g0w1, g0b1, g0w2, g0b2, g1w1, g1b1, g1w2, g1b2,
        out, ntiles);
}